// ExpMinProcessor_51951924412486
// MI455X (gfx1250) — compile-verified
//
#include <hip/hip_runtime.h>
#include <hip/hip_bf16.h>
#include <stdint.h>

// Top-p sampling, B=128 rows x V=128000 vocab.
// out[b, v] = +1e5 at winner token, -1e5 elsewhere.

#define NEGV (-100000.0f)
#define POSV (100000.0f)
#define TOPP 0.9f

#define NB 8192                  // histogram bins (32 KB LDS)
#define LMIN (-32.0f)
#define W1   (64.0f / (float)NB) // 0.0078125
#define INV1 ((float)NB / 64.0f) // 128
#define W2   (W1 / (float)NB)    // ~9.54e-7 logit resolution after refine
#define INV2 ((float)NB / W1)    // 1048576

#define CH 2048                  // TDM chunk (floats) = 8 KB per buffer

typedef unsigned int v4u __attribute__((ext_vector_type(4)));
typedef unsigned int v8u __attribute__((ext_vector_type(8)));

// ---------------------------------------------------------------------------
// TDM issue: 1-line tensor tile (len floats) from global -> LDS.
// D# per CDNA5 ISA ch.8: group0 {count=1, lds_addr, global_addr, type=2},
// group1 {data_size=2 (4B), tensor_dim0=len, tensor_dim1=1, tile_dim0=len,
//         tile_dim1=1, stride0=len}.  Issued once per wave (EXEC ignored).
// ---------------------------------------------------------------------------
__device__ __forceinline__ void tdm_load_1d(unsigned lds_off, const float* gptr,
                                            int len) {
    const unsigned long long ga = (unsigned long long)(size_t)gptr;
    v4u g0;
    g0.x = 1u;                                   // count=1, user descriptor
    g0.y = lds_off;                              // lds_addr (bytes)
    g0.z = (unsigned)ga;                         // global_addr[31:0]
    g0.w = ((unsigned)(ga >> 32) & 0x01FFFFFFu)  // global_addr[56:32]
           | 0x80000000u;                        // type=2 ("image")
    const unsigned d0 = (unsigned)len;
    v8u g1;
    g1.s0 = 0x00020000u;                         // wg_mask=0, data_size=4B
    g1.s1 = (d0 & 0xFFFFu) << 16;                // tensor_dim0[15:0]
    g1.s2 = ((d0 >> 16) & 0xFFFFu)               // tensor_dim0[31:16]
            | (1u << 16);                        // tensor_dim1 = 1
    g1.s3 = (d0 & 0xFFFFu) << 16;                // tile_dim0 = len
    g1.s4 = 1u;                                  // tile_dim1 = 1, tile_dim2 = 0
    g1.s5 = d0;                                  // tensor_dim0_stride[31:0]
    g1.s6 = 0u;                                  // stride0[47:32], stride1[15:0]
    g1.s7 = 0u;                                  // stride1[47:16]
    asm volatile("tensor_load_to_lds %0, %1"
                 :: "s"(g0), "s"(g1)
                 : "memory");
}

// ---------------------------------------------------------------------------
// Kernel 1: per-row two-level exp-mass histogram -> keep-threshold on logits.
// One workgroup (1024 threads) per row. Logits streamed by the Tensor Data
// Mover into a double-buffered LDS stage (TENSORcnt pipeline); the vector
// pipe only does exp + ds_add_f32 histogramming. Also inits the winner slot.
// ---------------------------------------------------------------------------
__global__ void __launch_bounds__(1024)
topp_hist_kernel(const float* __restrict__ logits, int V,
                 float* __restrict__ thr, unsigned long long* __restrict__ winner) {
    const int row = blockIdx.x;
    const int tid = threadIdx.x;

    __shared__ float h[NB];                       // 32 KB
    __shared__ float s1[1024];                    // 4 KB  (sums of 8 bins)
    __shared__ float s2[128];                     // 512 B (sums of 64 bins)
    __shared__ float red[1024];                   // 4 KB  (Z reduction)
    __shared__ __align__(16) float stage[2][CH];  // 16 KB TDM double buffer
    __shared__ float sCum;                        // mass strictly above bin b1
    __shared__ int   sB1;                         // crossing bin (level 1)

    for (int i = tid; i < NB; i += 1024) h[i] = 0.0f;
    __syncthreads();

    const float* lb = logits + (size_t)row * (size_t)V;
    const int nchunk = (V + CH - 1) / CH;
    const unsigned st0 = (unsigned)(size_t)&stage[0][0];
    const unsigned st1 = (unsigned)(size_t)&stage[1][0];

    // ================= phase 1: coarse histogram + Z ======================
    if (tid < 32) tdm_load_1d(st0, lb, min(CH, V));   // prologue (wave 0)

    float z = 0.0f;
    for (int c = 0; c < nchunk; ++c) {
        const int buf = c & 1;
        if (tid < 32) {
            const int cn = c + 1;
            if (cn < nchunk) {
                tdm_load_1d(buf ? st0 : st1, lb + (size_t)cn * CH,
                            min(CH, V - cn * CH));
                asm volatile("s_wait_tensorcnt 0x1" ::: "memory");
            } else {
                asm volatile("s_wait_tensorcnt 0x0" ::: "memory");
            }
        }
        __syncthreads();                       // chunk c landed, visible to all
        const int len = min(CH, V - c * CH);
        for (int i = tid; i < len; i += 1024) {
            float l = stage[buf][i];
            float e = __expf(fminf(l, 60.0f));
            z += e;
            int b = (int)((l - LMIN) * INV1);
            b = min(max(b, 0), NB - 1);
            atomicAdd(&h[b], e);               // ds_add_f32
        }
        __syncthreads();                       // buffer free for refill
    }

    red[tid] = z;
    __syncthreads();
    for (int s = 512; s > 0; s >>= 1) {
        if (tid < s) red[tid] += red[tid + s];
        __syncthreads();
    }
    // red[0] = Z (preserved through phase 2)

    {
        float a = 0.0f;
        #pragma unroll
        for (int j = 0; j < 8; ++j) a += h[tid * 8 + j];
        s1[tid] = a;
    }
    __syncthreads();
    if (tid < 128) {
        float a = 0.0f;
        #pragma unroll
        for (int j = 0; j < 8; ++j) a += s1[tid * 8 + j];
        s2[tid] = a;
    }
    __syncthreads();

    if (tid == 0) {
        const float target = TOPP * red[0];
        float acc = 0.0f;
        int k = 127;
        for (; k >= 0; --k) { float v = s2[k]; if (acc + v >= target) break; acc += v; }
        int b = 0;
        if (k >= 0) {
            int j = k * 8 + 7;
            for (; j >= k * 8; --j) { float v = s1[j]; if (acc + v >= target) break; acc += v; }
            if (j < k * 8) j = k * 8;
            b = j * 8 + 7;
            for (; b >= j * 8; --b) { float v = h[b]; if (acc + v >= target) break; acc += v; }
            if (b < j * 8) b = j * 8;
        }
        sB1 = b;
        sCum = acc;
    }
    __syncthreads();

    const int   b1      = sB1;
    const float lo1     = LMIN + (float)b1 * W1;
    const float target2 = TOPP * red[0] - sCum;  // mass still needed in bin b1

    // ============ phase 2: refine inside crossing bin (L2-hot) ============
    for (int i = tid; i < NB; i += 1024) h[i] = 0.0f;
    __syncthreads();

    if (tid < 32) tdm_load_1d(st0, lb, min(CH, V));   // prologue (wave 0)

    for (int c = 0; c < nchunk; ++c) {
        const int buf = c & 1;
        if (tid < 32) {
            const int cn = c + 1;
            if (cn < nchunk) {
                tdm_load_1d(buf ? st0 : st1, lb + (size_t)cn * CH,
                            min(CH, V - cn * CH));
                asm volatile("s_wait_tensorcnt 0x1" ::: "memory");
            } else {
                asm volatile("s_wait_tensorcnt 0x0" ::: "memory");
            }
        }
        __syncthreads();
        const int len = min(CH, V - c * CH);
        for (int i = tid; i < len; i += 1024) {
            float l = stage[buf][i];
            int bb = (int)((l - LMIN) * INV1);
            bb = min(max(bb, 0), NB - 1);
            if (bb == b1) {
                int b2 = (int)((l - lo1) * INV2);
                b2 = min(max(b2, 0), NB - 1);
                atomicAdd(&h[b2], __expf(fminf(l, 60.0f)));
            }
        }
        __syncthreads();
    }

    {
        float a = 0.0f;
        #pragma unroll
        for (int j = 0; j < 8; ++j) a += h[tid * 8 + j];
        s1[tid] = a;
    }
    __syncthreads();
    if (tid < 128) {
        float a = 0.0f;
        #pragma unroll
        for (int j = 0; j < 8; ++j) a += s1[tid * 8 + j];
        s2[tid] = a;
    }
    __syncthreads();

    if (tid == 0) {
        float acc = 0.0f;
        int k = 127;
        for (; k >= 0; --k) { float v = s2[k]; if (acc + v >= target2) break; acc += v; }
        int b = 0;
        if (k >= 0) {
            int j = k * 8 + 7;
            for (; j >= k * 8; --j) { float v = s1[j]; if (acc + v >= target2) break; acc += v; }
            if (j < k * 8) j = k * 8;
            b = j * 8 + 7;
            for (; b >= j * 8; --b) { float v = h[b]; if (acc + v >= target2) break; acc += v; }
            if (b < j * 8) b = j * 8;
        }
        thr[row] = lo1 + (float)b * W2;  // lower edge of crossing sub-bin
        winner[row] = 0xFFFFFFFFFFFFFFFFull;
    }
}

// ---------------------------------------------------------------------------
// Kernel 2: exponential-race argmin over the kept set.
// Streams logits+xi through LDS with global_load_async_to_lds_b128,
// software-pipelined with ASYNCcnt (double buffer: 2 batches in flight).
// score' = log(-log xi) - l  (monotone transform of -log(xi)/p).
// grid = (B rows, 25 column groups), 256 threads.
// ---------------------------------------------------------------------------
__device__ __forceinline__ void async_ld_b128(unsigned lds_off, unsigned byte_off,
                                              const float* base) {
    asm volatile("global_load_async_to_lds_b128 %0, %1, %2"
                 :: "v"(lds_off), "v"(byte_off),
                    "s"((unsigned long long)(size_t)base)
                 : "memory");
}

__global__ void __launch_bounds__(256)
topp_score_kernel(const float* __restrict__ logits, const float* __restrict__ xi,
                  const float* __restrict__ thr, unsigned long long* __restrict__ winner,
                  int V, int nbatch) {
    const int row = blockIdx.x;
    const int tid = threadIdx.x;
    const float* lb = logits + (size_t)row * (size_t)V;
    const float* xb = xi     + (size_t)row * (size_t)V;
    const float  t  = thr[row];

    __shared__ __align__(16) float bl[2][256 * 4];  // 8 KB logits staging
    __shared__ __align__(16) float bx[2][256 * 4];  // 8 KB xi staging
    __shared__ unsigned long long rmin[256];

    unsigned long long best = 0xFFFFFFFFFFFFFFFFull;

    int k = (int)blockIdx.y;              // batch index; batch = 1024 floats/block
    if (k < nbatch) {
        {   // prologue: batch k -> buffer 0
            unsigned off  = (unsigned)((k * 1024 + tid * 4) * 4);
            async_ld_b128((unsigned)(size_t)&bl[0][tid * 4], off, lb);
            async_ld_b128((unsigned)(size_t)&bx[0][tid * 4], off, xb);
        }
        int buf = 0;
        while (true) {
            const int kn = k + (int)gridDim.y;
            if (kn < nbatch) {
                // buffer buf^1 was fully consumed (values used) -> DS idle
                asm volatile("s_wait_dscnt 0" ::: "memory");
                unsigned off  = (unsigned)((kn * 1024 + tid * 4) * 4);
                async_ld_b128((unsigned)(size_t)&bl[buf ^ 1][tid * 4], off, lb);
                async_ld_b128((unsigned)(size_t)&bx[buf ^ 1][tid * 4], off, xb);
                asm volatile("s_wait_asynccnt 2" ::: "memory"); // batch k landed
            } else {
                asm volatile("s_wait_asynccnt 0" ::: "memory");
            }
            const float4 l4 = *(const float4*)&bl[buf][tid * 4];
            const float4 x4 = *(const float4*)&bx[buf][tid * 4];
            const int base = k * 1024 + tid * 4;
            #pragma unroll
            for (int j = 0; j < 4; ++j) {
                float l = (&l4.x)[j];
                if (l >= t) {
                    float x = (&x4.x)[j];
                    float s = __logf(-__logf(x)) - l;
                    unsigned kb = __float_as_uint(s);
                    kb = (kb & 0x80000000u) ? ~kb : (kb | 0x80000000u);
                    unsigned long long key =
                        ((unsigned long long)kb << 32) | (unsigned)(base + j);
                    best = key < best ? key : best;
                }
            }
            if (kn >= nbatch) break;
            k = kn;
            buf ^= 1;
        }
    }

    // generic tail (cols >= nbatch*1024); empty for V = 128000
    for (int c = nbatch * 1024 + (int)blockIdx.y * 256 + tid; c < V;
         c += (int)gridDim.y * 256) {
        float l = lb[c];
        if (l >= t) {
            float s = __logf(-__logf(xb[c])) - l;
            unsigned kb = __float_as_uint(s);
            kb = (kb & 0x80000000u) ? ~kb : (kb | 0x80000000u);
            unsigned long long key = ((unsigned long long)kb << 32) | (unsigned)c;
            best = key < best ? key : best;
        }
    }

    rmin[tid] = best;
    __syncthreads();
    for (int s = 128; s > 0; s >>= 1) {
        if (tid < s) {
            unsigned long long o = rmin[tid + s];
            if (o < rmin[tid]) rmin[tid] = o;
        }
        __syncthreads();
    }
    if (tid == 0) atomicMin(&winner[row], rmin[0]); // global_atomic_min_u64
}

// ---------------------------------------------------------------------------
// Kernel 3: one-hot fill, float4 stores. grid = (ceil(V/1024), B), 256 threads.
// ---------------------------------------------------------------------------
__global__ void __launch_bounds__(256)
topp_write_kernel(float* __restrict__ out,
                  const unsigned long long* __restrict__ winner, int V) {
    const int row = blockIdx.y;
    __shared__ unsigned sw;
    if (threadIdx.x == 0) sw = (unsigned)(winner[row] & 0xFFFFFFFFull);
    __syncthreads();
    const unsigned w = sw;

    float* ob = out + (size_t)row * (size_t)V;
    const int col = ((int)blockIdx.x * 256 + (int)threadIdx.x) * 4;
    if (col + 3 < V) {
        float4 o = make_float4(NEGV, NEGV, NEGV, NEGV);
        unsigned d = w - (unsigned)col;
        if (d < 4u) (&o.x)[d] = POSV;
        *(float4*)(ob + col) = o;
    } else {
        for (int c = col; c < V; ++c) ob[c] = ((unsigned)c == w) ? POSV : NEGV;
    }
}

// ---------------------------------------------------------------------------
extern "C" void kernel_launch(void* const* d_in, const int* in_sizes, int n_in,
                              void* d_out, int out_size, void* d_ws, size_t ws_size,
                              hipStream_t stream) {
    const float* logits = (const float*)d_in[1];
    const float* xi     = (const float*)d_in[2];
    // input_ids (d_in[0], int64) feeds only host-side seeding in the original.

    const int B = in_sizes[0] / 512;            // S = 512
    const int V = in_sizes[1] / B;              // 128000
    float* out = (float*)d_out;

    unsigned long long* winner = (unsigned long long*)d_ws;     // B * 8 bytes
    float* thr = (float*)(winner + B);                          // B * 4 bytes

    // 1) per-row threshold (TDM-streamed histogram) + winner init
    topp_hist_kernel<<<dim3(B), dim3(1024), 0, stream>>>(logits, V, thr, winner);

    // 2) async-pipelined race argmin
    const int nbatch = V / 1024;                // full 1024-float batches per row
    topp_score_kernel<<<dim3(B, 25), dim3(256), 0, stream>>>(logits, xi, thr,
                                                             winner, V, nbatch);

    // 3) one-hot fill
    topp_write_kernel<<<dim3((V + 1023) / 1024, B), dim3(256), 0, stream>>>(out,
                                                                            winner, V);
}